// GCN_63024350101829
// MI455X (gfx1250) — compile-verified
//
#include <hip/hip_runtime.h>
#include <hip/hip_bf16.h>

typedef float v2f __attribute__((ext_vector_type(2)));
typedef float v8f __attribute__((ext_vector_type(8)));

// ---------------------------------------------------------------------------
// Fast non-returning FP32 atomic add (lowers to global_atomic_add_f32).
__device__ __forceinline__ void atomAddF(float* p, float v) {
    (void)__hip_atomic_fetch_add(p, v, __ATOMIC_RELAXED, __HIP_MEMORY_SCOPE_AGENT);
}

// ---------------------------------------------------------------------------
__global__ void fill_kernel(float* __restrict__ p, float v, long long n) {
    long long t = (long long)blockIdx.x * blockDim.x + threadIdx.x;
    if (t < n) p[t] = v;
}

// pad x[N,6] -> xp[N,8] (cols 6,7 = 0); N*8 threads
__global__ void pad_x_kernel(const float* __restrict__ x, float* __restrict__ xp,
                             long long N) {
    long long t = (long long)blockIdx.x * blockDim.x + threadIdx.x;
    if (t >= N * 8) return;
    long long i = t >> 3;
    int f = (int)(t & 7);
    xp[t] = (f < 6) ? x[i * 6 + f] : 0.f;
}

// pad W1[6,32] -> Wp[8,32] (rows 6,7 = 0); one block of 256 threads
__global__ void pad_w1_kernel(const float* __restrict__ W1, float* __restrict__ Wp) {
    int t = threadIdx.x;
    int r = t >> 5;
    int c = t & 31;
    Wp[t] = (r < 6) ? W1[r * 32 + c] : 0.f;
}

// deg[col[e]] += 1  (self-loop "+1" is the fill value)
__global__ void degree_kernel(const int* __restrict__ col, float* __restrict__ deg,
                              long long E) {
    long long e = (long long)blockIdx.x * blockDim.x + threadIdx.x;
    if (e < E) atomAddF(&deg[col[e]], 1.0f);
}

// deg -> rsqrt(deg) in place (deg >= 1 guaranteed by self loops)
__global__ void rsqrt_kernel(float* __restrict__ d, long long n) {
    long long t = (long long)blockIdx.x * blockDim.x + threadIdx.x;
    if (t < n) d[t] = rsqrtf(fmaxf(d[t], 1.0f));
}

// norm[e] = dis[row[e]] * dis[col[e]]  (reused by all 3 layers)
__global__ void norm_kernel(const int* __restrict__ row, const int* __restrict__ col,
                            const float* __restrict__ dis, float* __restrict__ nrm,
                            long long E) {
    long long e = (long long)blockIdx.x * blockDim.x + threadIdx.x;
    if (e < E) nrm[e] = dis[row[e]] * dis[col[e]];
}

// ---------------------------------------------------------------------------
// hw[N,32] = h[N,K] @ W[K,32] via V_WMMA_F32_16X16X4_F32, K compile-time
// (8 or 32) so the loop fully unrolls with no bounds checks. Epilogue also
// writes agg = hw * dis^2 (self-loop contribution) straight from registers;
// interior tiles take an unguarded fast path (uniform wave branch), only the
// final partial tile pays per-row exec masking. One wave per 16-node tile;
// waves exit uniformly so EXEC is all-ones at every WMMA.
template <int K>
__global__ void gemm_wmma_kernel(const float* __restrict__ h,
                                 const float* __restrict__ W,
                                 const float* __restrict__ dis,
                                 float* __restrict__ hw,
                                 float* __restrict__ agg, int N) {
    int wave  = blockIdx.x * (blockDim.x >> 5) + (threadIdx.x >> 5);
    int lane  = threadIdx.x & 31;
    int node0 = wave * 16;
    if (node0 >= N) return;                  // uniform per wave

    const int n  = lane & 15;                // B/C/D column within 16-wide tile
    const int kh = (lane >> 4) * 2;          // half-wave K-pair select
    int row = node0 + (lane & 15);           // A row
    if (row >= N) row = N - 1;               // clamp: extra rows never stored
    const float* hrow = h + (size_t)row * K;

    v8f c0 = {0.f, 0.f, 0.f, 0.f, 0.f, 0.f, 0.f, 0.f};
    v8f c1 = {0.f, 0.f, 0.f, 0.f, 0.f, 0.f, 0.f, 0.f};

#pragma unroll
    for (int k0 = 0; k0 < K; k0 += 4) {
        v2f a = *(const v2f*)(hrow + k0 + kh);        // aligned b64 load
        v2f b0, b1;
        b0.x = W[(k0 + kh) * 32 + n];
        b0.y = W[(k0 + kh + 1) * 32 + n];
        b1.x = W[(k0 + kh) * 32 + 16 + n];
        b1.y = W[(k0 + kh + 1) * 32 + 16 + n];
        c0 = __builtin_amdgcn_wmma_f32_16x16x4_f32(false, a, false, b0,
                                                   (short)0, c0, false, false);
        c1 = __builtin_amdgcn_wmma_f32_16x16x4_f32(false, a, false, b1,
                                                   (short)0, c1, false, false);
    }

    // C/D layout: lane&15 = N col; lanes>=16 hold M+8; VGPR r = M within half
    const int mbase = (lane >> 4) * 8;
    if (node0 + 16 <= N) {
        // fast path: full tile, no per-row guards
        float s[8];
#pragma unroll
        for (int r = 0; r < 8; ++r) {
            float d = dis[node0 + mbase + r];
            s[r] = d * d;
        }
#pragma unroll
        for (int r = 0; r < 8; ++r) {
            size_t base = (size_t)(node0 + mbase + r) * 32 + n;
            float v0 = c0[r], v1 = c1[r];
            hw[base]           = v0;
            hw[base + 16]      = v1;
            agg[base]          = v0 * s[r];
            agg[base + 16]     = v1 * s[r];
        }
    } else {
        // tail tile: per-row guard
#pragma unroll
        for (int r = 0; r < 8; ++r) {
            int mm = node0 + mbase + r;
            if (mm < N) {
                float d = dis[mm];
                float sc = d * d;
                size_t base = (size_t)mm * 32 + n;
                float v0 = c0[r], v1 = c1[r];
                hw[base]       = v0;
                hw[base + 16]  = v1;
                agg[base]      = v0 * sc;
                agg[base + 16] = v1 * sc;
            }
        }
    }
}

// ---------------------------------------------------------------------------
// agg[col,f] += hw[row,f] * norm[e]. One wave handles 32 edges: edge data is
// loaded coalesced (one lane per edge), then each edge is broadcast via shfl
// while the 32 lanes cover the 32 features.
__global__ void edge_scatter_kernel(const int* __restrict__ row,
                                    const int* __restrict__ col,
                                    const float* __restrict__ nrm,
                                    const float* __restrict__ hw,
                                    float* __restrict__ agg, long long E) {
    int lane = threadIdx.x & 31;
    long long wid = (((long long)blockIdx.x * blockDim.x) + threadIdx.x) >> 5;
    long long e0 = wid * 32;
    if (e0 >= E) return;                     // uniform per wave
    long long e = e0 + lane;
    bool ok = e < E;
    int   r = ok ? row[e] : 0;
    int   c = ok ? col[e] : 0;
    float w = ok ? nrm[e] : 0.f;
    int cnt = (int)((E - e0) < 32 ? (E - e0) : 32);   // uniform
    for (int j = 0; j < cnt; ++j) {
        int   rj = __shfl(r, j);
        int   cj = __shfl(c, j);
        float wj = __shfl(w, j);
        atomAddF(&agg[(size_t)cj * 32 + lane], hw[(size_t)rj * 32 + lane] * wj);
    }
}

// h[i,f] = tanh(agg[i,f] + b[f]), float4-vectorized (N*8 threads)
__global__ void tanh_bias_kernel(const float* __restrict__ agg,
                                 const float* __restrict__ b,
                                 float* __restrict__ h, long long N) {
    long long t = (long long)blockIdx.x * blockDim.x + threadIdx.x;
    if (t >= N * 8) return;
    int q = (int)(t & 7);
    float4 a  = ((const float4*)agg)[t];
    float4 bb = ((const float4*)b)[q];
    float4 o;
    o.x = tanhf(a.x + bb.x);
    o.y = tanhf(a.y + bb.y);
    o.z = tanhf(a.z + bb.z);
    o.w = tanhf(a.w + bb.w);
    ((float4*)h)[t] = o;
}

// pool[batch[i],f] += h[i,f]; cnt[batch[i]] += 1 (once per node)
__global__ void pool_kernel(const float* __restrict__ h,
                            const int* __restrict__ batch,
                            float* __restrict__ pool, float* __restrict__ cnt,
                            long long N) {
    long long t = (long long)blockIdx.x * blockDim.x + threadIdx.x;
    if (t >= N * 32) return;
    long long i = t >> 5;
    int f = (int)(t & 31);
    int g = batch[i];
    atomAddF(&pool[(size_t)g * 32 + f], h[t]);
    if (f == 0) atomAddF(&cnt[g], 1.0f);
}

// out[g,c] = (pool[g,:]/max(cnt,1)) @ Wc + bc   (G*6 threads; tiny)
__global__ void classify_kernel(const float* __restrict__ pool,
                                const float* __restrict__ cnt,
                                const float* __restrict__ Wc,
                                const float* __restrict__ bc,
                                float* __restrict__ out, int G) {
    int t = blockIdx.x * blockDim.x + threadIdx.x;
    if (t >= G * 6) return;
    int g = t / 6, c = t % 6;
    float inv = 1.0f / fmaxf(cnt[g], 1.0f);
    float acc = bc[c];
    const float* pg = pool + (size_t)g * 32;
    for (int f = 0; f < 32; ++f) acc += pg[f] * inv * Wc[f * 6 + c];
    out[t] = acc;
}

// ---------------------------------------------------------------------------
extern "C" void kernel_launch(void* const* d_in, const int* in_sizes, int n_in,
                              void* d_out, int out_size, void* d_ws, size_t ws_size,
                              hipStream_t stream) {
    const float* x     = (const float*)d_in[0];
    const int*   ei    = (const int*)d_in[1];   // [2,E]: [0]=src(row), [1]=dst(col)
    const int*   batch = (const int*)d_in[2];
    const float* W1 = (const float*)d_in[3];
    const float* b1 = (const float*)d_in[4];
    const float* W2 = (const float*)d_in[5];
    const float* b2 = (const float*)d_in[6];
    const float* W3 = (const float*)d_in[7];
    const float* b3 = (const float*)d_in[8];
    const float* Wc = (const float*)d_in[9];
    const float* bc = (const float*)d_in[10];
    float* out = (float*)d_out;

    const long long N = in_sizes[0] / 6;
    const long long E = in_sizes[1] / 2;
    const int       G = out_size / 6;

    const int* rowIdx = ei;        // sources
    const int* colIdx = ei + E;    // targets

    // ws layout (floats):
    // dis[N] | xpad[8N] | hw[32N] | agg[32N] | h[32N] | pool[32G] | cnt[G] | W1pad[256] | norm[E]
    float* ws    = (float*)d_ws;
    float* dis   = ws;
    float* xpad  = dis + N;
    float* hw    = xpad + 8 * N;
    float* agg   = hw + 32 * N;
    float* h     = agg + 32 * N;
    float* pool  = h + 32 * N;
    float* cnt   = pool + 32 * (long long)G;
    float* W1pad = cnt + G;
    float* nrm   = W1pad + 256;

    const int BS = 256;
    auto blocks = [](long long work, int bs) { return (int)((work + bs - 1) / bs); };
    const int gemmBlocks = blocks(((N + 15) / 16) * 32, BS);      // 1 wave / 16 nodes
    const int edgeBlocks = blocks(((E + 31) / 32) * 32, BS);      // 1 wave / 32 edges

    // ---- setup: padding, degree, normalization ------------------------------
    fill_kernel<<<blocks(N, BS), BS, 0, stream>>>(dis, 1.0f, N);                 // self loop
    fill_kernel<<<blocks(33LL * G, BS), BS, 0, stream>>>(pool, 0.0f, 33LL * G);  // pool+cnt
    pad_x_kernel<<<blocks(8 * N, BS), BS, 0, stream>>>(x, xpad, N);
    pad_w1_kernel<<<1, 256, 0, stream>>>(W1, W1pad);
    degree_kernel<<<blocks(E, BS), BS, 0, stream>>>(colIdx, dis, E);
    rsqrt_kernel<<<blocks(N, BS), BS, 0, stream>>>(dis, N);
    norm_kernel<<<blocks(E, BS), BS, 0, stream>>>(rowIdx, colIdx, dis, nrm, E);

    // ---- layer 1: xpad[N,8] -> h[N,32] --------------------------------------
    gemm_wmma_kernel<8><<<gemmBlocks, BS, 0, stream>>>(xpad, W1pad, dis, hw, agg, (int)N);
    edge_scatter_kernel<<<edgeBlocks, BS, 0, stream>>>(rowIdx, colIdx, nrm, hw, agg, E);
    tanh_bias_kernel<<<blocks(8 * N, BS), BS, 0, stream>>>(agg, b1, h, N);

    // ---- layer 2 ------------------------------------------------------------
    gemm_wmma_kernel<32><<<gemmBlocks, BS, 0, stream>>>(h, W2, dis, hw, agg, (int)N);
    edge_scatter_kernel<<<edgeBlocks, BS, 0, stream>>>(rowIdx, colIdx, nrm, hw, agg, E);
    tanh_bias_kernel<<<blocks(8 * N, BS), BS, 0, stream>>>(agg, b2, h, N);

    // ---- layer 3 ------------------------------------------------------------
    gemm_wmma_kernel<32><<<gemmBlocks, BS, 0, stream>>>(h, W3, dis, hw, agg, (int)N);
    edge_scatter_kernel<<<edgeBlocks, BS, 0, stream>>>(rowIdx, colIdx, nrm, hw, agg, E);
    tanh_bias_kernel<<<blocks(8 * N, BS), BS, 0, stream>>>(agg, b3, h, N);

    // ---- global mean pool + classifier --------------------------------------
    pool_kernel<<<blocks(32 * N, BS), BS, 0, stream>>>(h, batch, pool, cnt, N);
    classify_kernel<<<blocks(6LL * G, BS), BS, 0, stream>>>(pool, cnt, Wc, bc, out, G);
}